// ClassificationHead_63307817943776
// MI455X (gfx1250) — compile-verified
//
#include <hip/hip_runtime.h>

typedef __attribute__((ext_vector_type(2))) float v2f;
typedef __attribute__((ext_vector_type(8))) float v8f;

#define NTASKS 1024
#define NS 75          // support rows
#define NQ 75          // query rows
#define DIM 1024
#define NW 5           // n_way
#define LAMBDA_REG 100.0f

#define DC 64          // D-chunk width streamed through LDS
#define SP 68          // LDS row stride (floats) for S/Q chunks: 16B-aligned rows, conflict-free gather
#define KSTR 81        // LDS row stride (floats) for K / Ksq (80x80 padded tiles)
#define TPB 320        // 10 waves (wave32): wave = (tm in 0..4) x (h in 0..1)

// CDNA5 async global->LDS copy: 16B per lane, bypasses VGPRs, tracked by ASYNCcnt.
__device__ __forceinline__ void async_copy16(const float* gptr, float* lptr) {
    unsigned lds_off = (unsigned)(unsigned long long)lptr;   // LDS aperture: addr[31:0] = byte offset
    asm volatile("global_load_async_to_lds_b128 %0, %1, off"
                 :: "v"(lds_off), "v"(gptr) : "memory");
}
__device__ __forceinline__ void wait_async_all() {
    asm volatile("s_wait_asynccnt 0x0" ::: "memory");
}

__global__ __launch_bounds__(TPB)
void classification_head_kernel(const float* __restrict__ query,
                                const float* __restrict__ support,
                                const int*   __restrict__ labels,
                                float* __restrict__ out)
{
    __shared__ float sS[80 * SP];        // support chunk, rows 75..79 kept zero
    __shared__ float sQ[80 * SP];        // query chunk, rows 75..79 kept zero
    __shared__ float sK[80 * KSTR];      // gram matrix -> Cholesky factor (lower tri)
    __shared__ float sKsq[80 * KSTR];    // support-query compatibility [s][q]
    __shared__ float sX[80 * NW];        // qp solution (2 * K^-1 Y)
    __shared__ float sLkk;

    const int task = blockIdx.x;
    const int tid  = threadIdx.x;
    const int lane = tid & 31;
    const int wv   = tid >> 5;           // 0..9
    const int tm   = wv % 5;             // row block of C this wave owns
    const int h    = wv / 5;             // 0: K tiles (B = S), 1: Ksq tiles (B = Q)

    const float* Sg = support + (size_t)task * NS * DIM;
    const float* Qg = query   + (size_t)task * NQ * DIM;

    // zero the pad rows once; chunk staging only ever writes rows < 75
    for (int idx = tid; idx < 5 * SP; idx += TPB) {
        sS[75 * SP + idx] = 0.0f;
        sQ[75 * SP + idx] = 0.0f;
    }

    // fragment gather (identical pattern for A 16x4 and B 4x16 of a gram product):
    // element(lane, j) = Rows[16*tile + (lane&15)][kb + 2*(lane>>4) + j]
    const int mrow = lane & 15;
    const int koff = (lane >> 4) << 1;

    const float* Abase = &sS[(tm * 16 + mrow) * SP + koff];
    const float* Bmat  = h ? sQ : sS;
    const float* Bbase = &Bmat[mrow * SP + koff];

    v8f acc[5];
#pragma unroll
    for (int j = 0; j < 5; ++j) acc[j] = 0.0f;

    for (int c0 = 0; c0 < DIM; c0 += DC) {
        // ---- async-stage S and Q chunks into LDS (16B per lane, no VGPR roundtrip) ----
        for (int idx = tid; idx < NS * (DC / 4); idx += TPB) {
            int r  = idx / (DC / 4);
            int c4 = (idx % (DC / 4)) * 4;
            async_copy16(Sg + (size_t)r * DIM + c0 + c4, &sS[r * SP + c4]);
            async_copy16(Qg + (size_t)r * DIM + c0 + c4, &sQ[r * SP + c4]);
        }
        if (c0 + DC < DIM && tid < NS) {        // prefetch next chunk into L2
            __builtin_prefetch(Sg + (size_t)tid * DIM + c0 + DC);
            __builtin_prefetch(Qg + (size_t)tid * DIM + c0 + DC);
        }
        wait_async_all();
        __syncthreads();

        // ---- WMMA: one shared A fragment feeds 5 tiles per k-step ----
        for (int kb = 0; kb < DC; kb += 4) {
            v2f a = *(const v2f*)(Abase + kb);
#pragma unroll
            for (int j = 0; j < 5; ++j) {
                v2f b = *(const v2f*)(Bbase + j * 16 * SP + kb);
                acc[j] = __builtin_amdgcn_wmma_f32_16x16x4_f32(
                    false, a, false, b, (short)0, acc[j], false, false);
            }
        }
        __syncthreads();
    }

    // ---- spill accumulators: C layout M = v + 8*(lane>>4), N = lane&15 ----
    {
        float* Cb = h ? sKsq : sK;
        const int rbase = tm * 16 + 8 * (lane >> 4);
#pragma unroll
        for (int j = 0; j < 5; ++j) {
            const int col = j * 16 + (lane & 15);
#pragma unroll
            for (int v = 0; v < 8; ++v)
                Cb[(size_t)(rbase + v) * KSTR + col] = acc[j][v];
        }
    }
    __syncthreads();

    // ---- ridge regularization ----
    for (int i = tid; i < NS; i += TPB) sK[i * KSTR + i] += LAMBDA_REG;
    __syncthreads();

    // ---- right-looking Cholesky of K (lower triangle becomes L) ----
    for (int k = 0; k < NS; ++k) {
        if (tid == 0) {
            float d = sqrtf(sK[k * KSTR + k]);
            sK[k * KSTR + k] = d;
            sLkk = d;
        }
        __syncthreads();
        const float inv = 1.0f / sLkk;
        for (int i = k + 1 + tid; i < NS; i += TPB) sK[i * KSTR + k] *= inv;
        __syncthreads();
        const int m = NS - 1 - k;
        for (int idx = tid; idx < m * m; idx += TPB) {
            const int i = k + 1 + idx / m;
            const int j = k + 1 + idx % m;
            sK[i * KSTR + j] -= sK[i * KSTR + k] * sK[j * KSTR + k];
        }
        __syncthreads();
    }

    // ---- triangular solves: x = 2 * K^{-1} onehot(labels), one thread per class ----
    if (tid < NW) {
        const int w = tid;
        const int* lab = labels + (size_t)task * NS;
        // forward: L z = 2*y   (fold the factor 2 into the RHS)
        for (int i = 0; i < NS; ++i) {
            float t = (lab[i] == w) ? 2.0f : 0.0f;
            for (int j = 0; j < i; ++j) t -= sK[i * KSTR + j] * sX[j * NW + w];
            sX[i * NW + w] = t / sK[i * KSTR + i];
        }
        // backward: L^T x = z  (in place)
        for (int i = NS - 1; i >= 0; --i) {
            float t = sX[i * NW + w];
            for (int j = i + 1; j < NS; ++j) t -= sK[j * KSTR + i] * sX[j * NW + w];
            sX[i * NW + w] = t / sK[i * KSTR + i];
        }
    }
    __syncthreads();

    // ---- logits[q][w] = sum_s Ksq[s][q] * x[s][w] ----
    float* Og = out + (size_t)task * NQ * NW;
    for (int idx = tid; idx < NQ * NW; idx += TPB) {
        const int q = idx / NW;
        const int w = idx % NW;
        float t = 0.0f;
        for (int s = 0; s < NS; ++s) t += sKsq[s * KSTR + q] * sX[s * NW + w];
        Og[idx] = t;
    }
}

extern "C" void kernel_launch(void* const* d_in, const int* in_sizes, int n_in,
                              void* d_out, int out_size, void* d_ws, size_t ws_size,
                              hipStream_t stream) {
    (void)in_sizes; (void)n_in; (void)out_size; (void)d_ws; (void)ws_size;
    const float* query   = (const float*)d_in[0];
    const float* support = (const float*)d_in[1];
    const int*   labels  = (const int*)d_in[2];
    float* out = (float*)d_out;
    classification_head_kernel<<<dim3(NTASKS), dim3(TPB), 0, stream>>>(
        query, support, labels, out);
}